// FeatureExtractor_2199023256259
// MI455X (gfx1250) — compile-verified
//
#include <hip/hip_runtime.h>

typedef __attribute__((ext_vector_type(2))) float v2f;
typedef __attribute__((ext_vector_type(8))) float v8f;

#define NSEQ   16384      // B*Ns = 32*512
#define TSTEPS 128
#define CIN    25
#define PDIM   20
#define HDIM   25
#define GDIM   75         // 3*H
#define WPB    2          // waves per block (wave32)
#define RPW    16         // rows (sequences) per wave = WMMA M
#define BLOCK_THREADS (WPB*32)
#define AOPS   13         // async b32 ops per x tile: ceil(16*25/32)

// LDS geometry (padded so every WMMA operand access is unconditional)
#define XS   28           // x tile col stride (25 -> 28)
#define W1R  32           // W1 rows padded 20 -> 32
#define WGR  80           // Wih/Whh rows padded 75 -> 80
#define PKS  20           // Wih K stride (= PDIM)
#define HS   28           // h / Whh col stride (25 -> 28)
#define PBS  32           // p tile col stride
#define GBS  80           // gate buffer col stride
#define HNS  32           // gh n-part buffer: gate cols 48..79

__device__ __forceinline__ v8f wmma4(v2f a, v2f b, v8f c) {
#if defined(__AMDGCN__) && __has_builtin(__builtin_amdgcn_wmma_f32_16x16x4_f32)
  return __builtin_amdgcn_wmma_f32_16x16x4_f32(false, a, false, b, (short)0, c,
                                               false, false);
#else
  return c; // host-pass placeholder
#endif
}

__device__ __forceinline__ float fast_rcp(float v) {
#if __has_builtin(__builtin_amdgcn_rcpf)
  return __builtin_amdgcn_rcpf(v);
#else
  return 1.f / v;
#endif
}
__device__ __forceinline__ float fast_tanh(float v) {
#if __has_builtin(__builtin_amdgcn_tanhf)
  return __builtin_amdgcn_tanhf(v);
#else
  return tanhf(v);
#endif
}
__device__ __forceinline__ float fast_sigmoid(float v) {
  return fast_rcp(1.f + __expf(-v));
}

__global__ __launch_bounds__(BLOCK_THREADS, 1)
void gru_fused_kernel(const float* __restrict__ x,
                      const float* __restrict__ W1,  const float* __restrict__ b1,
                      const float* __restrict__ Wih, const float* __restrict__ Whh,
                      const float* __restrict__ bih, const float* __restrict__ bhh,
                      float* __restrict__ out)
{
  __shared__ float sW1 [W1R * XS];
  __shared__ float sWih[WGR * PKS];
  __shared__ float sWhh[WGR * HS];
  __shared__ float sB1p[32];
  __shared__ float sBrz[50];
  __shared__ float sBn [32];
  __shared__ float sX [WPB][2][RPW * XS];   // double-buffered async x tiles
  __shared__ float sP [WPB][RPW * PBS];
  __shared__ float sG [WPB][RPW * GBS];
  __shared__ float sHn[WPB][RPW * HNS];
  __shared__ float sH [WPB][RPW * HS];

  const int tid  = threadIdx.x;
  const int wave = tid >> 5;
  const int lane = tid & 31;

  // ---- cooperative weight preload + zero padding (once) ----
  for (int i = tid; i < W1R * XS; i += BLOCK_THREADS) {
    int r = i / XS, c = i - r * XS;
    sW1[i] = (r < PDIM && c < CIN) ? W1[r * CIN + c] : 0.f;
  }
  for (int i = tid; i < WGR * PKS; i += BLOCK_THREADS) {
    int r = i / PKS, c = i - r * PKS;
    sWih[i] = (r < GDIM) ? Wih[r * PDIM + c] : 0.f;
  }
  for (int i = tid; i < WGR * HS; i += BLOCK_THREADS) {
    int r = i / HS, c = i - r * HS;
    sWhh[i] = (r < GDIM && c < HDIM) ? Whh[r * HDIM + c] : 0.f;
  }
  for (int i = tid; i < 32; i += BLOCK_THREADS) sB1p[i] = (i < PDIM) ? b1[i] : 0.f;
  for (int i = tid; i < 50; i += BLOCK_THREADS) sBrz[i] = bih[i] + bhh[i];
  for (int i = tid; i < 32; i += BLOCK_THREADS)
    sBn[i] = (i < HDIM) ? (bih[50 + i] + bhh[50 + i]) : 0.f;
  for (int i = tid; i < WPB * RPW * HS; i += BLOCK_THREADS) (&sH[0][0])[i] = 0.f;
  __syncthreads();

  const int n0 = blockIdx.x * (WPB * RPW) + wave * RPW;
  float* xb0 = sX[wave][0];
  float* xb1 = sX[wave][1];
  float* pb  = sP[wave];
  float* gb  = sG[wave];
  float* hnb = sHn[wave];
  float* hb  = sH[wave];

  // zero the x pad columns (25..27) of both buffers once: 32 rows, 1 per lane
  {
    float* bx = (lane < RPW) ? xb0 : xb1;
    int r = lane & 15;
    bx[r * XS + 25] = 0.f; bx[r * XS + 26] = 0.f; bx[r * XS + 27] = 0.f;
  }

  // WMMA f32 16x16x4 lane mappings (ISA 7.12.2)
  const int mA    = lane & 15;
  const int kHalf = (lane >> 4) << 1;
  const int nB    = lane & 15;
  const int dRow0 = (lane < 16) ? 0 : 8;
  const int dCol  = lane & 15;

  // ---- hoist ALL loop-invariant B-fragments + biases into registers ----
  v2f bW1f[2][XS / 4], bWhhf[5][HS / 4], bWihf[5][PKS / 4];
#pragma unroll
  for (int pt = 0; pt < 2; ++pt)
#pragma unroll
    for (int kk = 0; kk < XS / 4; ++kk) {
      bW1f[pt][kk].x = sW1[(pt * 16 + nB) * XS + kk * 4 + kHalf];
      bW1f[pt][kk].y = sW1[(pt * 16 + nB) * XS + kk * 4 + kHalf + 1];
    }
#pragma unroll
  for (int gt = 0; gt < 5; ++gt) {
#pragma unroll
    for (int kk = 0; kk < HS / 4; ++kk) {
      bWhhf[gt][kk].x = sWhh[(gt * 16 + nB) * HS + kk * 4 + kHalf];
      bWhhf[gt][kk].y = sWhh[(gt * 16 + nB) * HS + kk * 4 + kHalf + 1];
    }
#pragma unroll
    for (int kk = 0; kk < PKS / 4; ++kk) {
      bWihf[gt][kk].x = sWih[(gt * 16 + nB) * PKS + kk * 4 + kHalf];
      bWihf[gt][kk].y = sWih[(gt * 16 + nB) * PKS + kk * 4 + kHalf + 1];
    }
  }
  float biasP[2] = { sB1p[dCol], sB1p[16 + dCol] };
  float gbias[5];
  gbias[0] = sBrz[dCol];
  gbias[1] = sBrz[16 + dCol];
  gbias[2] = sBrz[32 + dCol];
  gbias[3] = (dCol < 2) ? sBrz[48 + dCol] : 0.f;
  gbias[4] = 0.f;
  float bHn[2];
  bHn[0] = bhh[48 + dCol];
  bHn[1] = (64 + dCol < GDIM) ? bhh[64 + dCol] : 0.f;

  // ---- async x-tile pipeline setup: element e = i*32+lane of the 16x25 tile
  const float* xbase = x + (long)n0 * TSTEPS * CIN;   // uniform SADDR
  int      boff [AOPS];   // per-lane global byte offset (advances 100B per t)
  unsigned loffB[AOPS];   // per-lane LDS byte offset within a tile buffer
#pragma unroll
  for (int i = 0; i < AOPS; ++i) {
    int e = i * 32 + lane;
    int row = (e < RPW * CIN) ? (e / CIN) : 0;
    int col = e - row * CIN;
    if (e >= RPW * CIN) col = 0;
    boff[i]  = ((row * TSTEPS) * CIN + col) * 4;
    loffB[i] = (unsigned)(row * XS + col) * 4;
  }
  const unsigned xbB[2] = { (unsigned)(uintptr_t)xb0, (unsigned)(uintptr_t)xb1 };

  auto issue_tile = [&](unsigned ldsBase) {
#pragma unroll
    for (int i = 0; i < AOPS; ++i) {
      if (i < AOPS - 1 || lane < 16) {   // tail op: only 16 lanes carry data
#if defined(__AMDGCN__)
        asm volatile("global_load_async_to_lds_b32 %0, %1, %2"
                     :: "v"(ldsBase + loffB[i]), "v"(boff[i]), "s"(xbase)
                     : "memory");
#endif
      }
      boff[i] += CIN * 4;                // advance one timestep
    }
  };

  issue_tile(xbB[0]);                    // prefetch t = 0

  // lane-constant elementwise mapping: 32 lanes = 16 rows x 2 halves
  const int m    = lane >> 1;
  const int half = lane & 1;
  float*       gRow  = gb  + m * GBS + half * 13;
  float*       hnRow = hnb + m * HNS + half * 13 + 2;
  float*       hRow  = hb  + m * HS  + half * 13;
  const float* bn    = sBn + half * 13;

  for (int t = 0; t < TSTEPS; ++t) {
    // issue DMA for t+1 into the other buffer, then wait for t's tile
    if (t + 1 < TSTEPS) {
      issue_tile(xbB[(t + 1) & 1]);
#if defined(__AMDGCN__)
      asm volatile("s_wait_asynccnt 0xd" ::: "memory");   // t batch landed
#endif
    } else {
#if defined(__AMDGCN__)
      asm volatile("s_wait_asynccnt 0x0" ::: "memory");
#endif
    }
    const float* xbc = (t & 1) ? xb1 : xb0;

    // ---- projection p = LeakyReLU(x @ W1^T + b1) ----
    {
      v2f ax[XS / 4];
#pragma unroll
      for (int kk = 0; kk < XS / 4; ++kk) {
        ax[kk].x = xbc[mA * XS + kk * 4 + kHalf];
        ax[kk].y = xbc[mA * XS + kk * 4 + kHalf + 1];
      }
#pragma unroll
      for (int pt = 0; pt < 2; ++pt) {
        v8f acc = {};
#pragma unroll
        for (int kk = 0; kk < XS / 4; ++kk) acc = wmma4(ax[kk], bW1f[pt][kk], acc);
#pragma unroll
        for (int r = 0; r < 8; ++r) {
          float v = acc[r] + biasP[pt];
          v = (v >= 0.f) ? v : 0.01f * v;
          pb[(dRow0 + r) * PBS + pt * 16 + dCol] = v;
        }
      }
    }

    // ---- gates: acc = bias_rz + h@Whh^T + p@Wih^T ; 5 N-tiles ----
    {
      v2f ah[HS / 4], ap[PKS / 4];
#pragma unroll
      for (int kk = 0; kk < HS / 4; ++kk) {
        ah[kk].x = hb[mA * HS + kk * 4 + kHalf];
        ah[kk].y = hb[mA * HS + kk * 4 + kHalf + 1];
      }
#pragma unroll
      for (int kk = 0; kk < PKS / 4; ++kk) {
        ap[kk].x = pb[mA * PBS + kk * 4 + kHalf];
        ap[kk].y = pb[mA * PBS + kk * 4 + kHalf + 1];
      }
#pragma unroll
      for (int gt = 0; gt < 5; ++gt) {
        v8f acc;
#pragma unroll
        for (int r = 0; r < 8; ++r) acc[r] = gbias[gt];
#pragma unroll
        for (int kk = 0; kk < HS / 4; ++kk) acc = wmma4(ah[kk], bWhhf[gt][kk], acc);
        if (gt >= 3) {
#pragma unroll
          for (int r = 0; r < 8; ++r)
            hnb[(dRow0 + r) * HNS + (gt - 3) * 16 + dCol] = acc[r] + bHn[gt - 3];
        }
#pragma unroll
        for (int kk = 0; kk < PKS / 4; ++kk) acc = wmma4(ap[kk], bWihf[gt][kk], acc);
#pragma unroll
        for (int r = 0; r < 8; ++r)
          gb[(dRow0 + r) * GBS + gt * 16 + dCol] = acc[r];
      }
    }

    // ---- elementwise GRU update ----
#pragma unroll
    for (int u = 0; u < 13; ++u) {
      if (u < 12 || half == 0) {
        float rg = fast_sigmoid(gRow[u]);
        float zg = fast_sigmoid(gRow[25 + u]);
        float hn = hnRow[u];
        float xn = gRow[50 + u] - hn + bn[u];
        float ng = fast_tanh(xn + rg * hn);
        float hp = hRow[u];
        hRow[u] = ng + zg * (hp - ng);
      }
    }
  }

  // ---- emit h_T ----
#pragma unroll
  for (int u = 0; u < 13; ++u)
    if (u < 12 || half == 0)
      out[(long)(n0 + m) * HDIM + half * 13 + u] = hRow[u];
}

extern "C" void kernel_launch(void* const* d_in, const int* in_sizes, int n_in,
                              void* d_out, int out_size, void* d_ws, size_t ws_size,
                              hipStream_t stream) {
  (void)in_sizes; (void)n_in; (void)out_size; (void)d_ws; (void)ws_size;
  const float* x   = (const float*)d_in[0];
  const float* W1  = (const float*)d_in[1];
  const float* b1  = (const float*)d_in[2];
  const float* Wih = (const float*)d_in[3];
  const float* Whh = (const float*)d_in[4];
  const float* bih = (const float*)d_in[5];
  const float* bhh = (const float*)d_in[6];
  float* out = (float*)d_out;

  dim3 grid(NSEQ / (WPB * RPW));   // 512 blocks * 2 waves * 16 rows
  dim3 block(BLOCK_THREADS);
  hipLaunchKernelGGL(gru_fused_kernel, grid, block, 0, stream,
                     x, W1, b1, Wih, Whh, bih, bhh, out);
}